// ConvCaps_87814901334039
// MI455X (gfx1250) — compile-verified
//
#include <hip/hip_runtime.h>
#include <hip/hip_bf16.h>

// ConvCaps EM-routing for MI455X (gfx1250), wave32.
// Workspace requirement: ~48.15M floats (~193 MB).
//   E (votes, f32, [b][k=288][flat=18432])  : 42,467,328
//   R, RH/PH (shared) ([b][288][1152] each) :  2,654,208 * 2
//   MU, SG ([b][18432] each)                :    147,456 * 2
//   SR, ACT, PT ([b][1152] each)            :      9,216 * 3
//   S ([b][32][14][14])                     :     50,176

typedef float v2f __attribute__((ext_vector_type(2)));
typedef float v8f __attribute__((ext_vector_type(8)));

#define NB    8
#define NK    288      // (u,x,y) = 32*3*3
#define NT    1152     // (i,j,c) = 6*6*32
#define NF    18432    // 16 * 1152 (flat vote block per (b,k))
#define CH_X  544      // 512 pose channels + 32 act channels
#define HH    14

// ---------------------------------------------------------------------------
// Kernel 1: votes via V_WMMA_F32_16X16X4_F32.
// Per block: fixed (b, k=(u,xk,yk)). GEMM: M=(i,j,q)=144 rows (9 tiles),
// N=(c,p)=128 cols (8 tiles, one per wave), K=m=4 (single WMMA per tile).
// Division-free inner loop: q is lane-constant, (i,j) advanced by carry-add,
// D fragment stored as two aligned float4s.
// ---------------------------------------------------------------------------
__global__ __launch_bounds__(256) void capsvote_wmma(
    const float* __restrict__ xin, const float* __restrict__ Wt,
    float* __restrict__ E)
{
  const int bid = blockIdx.x;           // 0..2303  (= b*288 + k)
  const int b  = bid / NK;
  const int k  = bid % NK;
  const int u  = k / 9;
  const int xk = (k % 9) / 3;
  const int yk = k % 3;

  const int wave = threadIdx.x >> 5;    // N-tile id (0..7)
  const int lane = threadIdx.x & 31;
  const int half = lane >> 4;           // lane half selects K split
  const int l16  = lane & 15;

  // B fragment: B[m, nu] = W[xk][yk][u][c][p][m], nu = wave*16 + l16
  const int nu = wave * 16 + l16;
  const int c  = nu >> 2;
  const int p  = nu & 3;
  v2f bfrag;
#pragma unroll
  for (int r = 0; r < 2; ++r) {
    const int m = half * 2 + r;
    bfrag[r] = Wt[((((xk * 3 + yk) * 32 + u) * 32 + c) * 4 + p) * 4 + m];
  }

  // A addressing: rho = Mt*16 + l16 = (i*6+j)*4 + q; q = l16 & 3 fixed.
  const int q = l16 & 3;
  int j = l16 >> 2;                     // initial ij in [0,4) -> i=0, j=ij
  // pix = (2*i + xk)*14 + (2*j + yk), maintained incrementally
  int pix = xk * HH + yk + 2 * j;
  int baseA[2];
#pragma unroll
  for (int r = 0; r < 2; ++r) {
    const int m  = half * 2 + r;
    const int ch = m * 128 + q * 32 + u;
    baseA[r] = (b * CH_X + ch) * (HH * HH);
  }

  const size_t ebase = (size_t)(b * NK + k) * NF;
  // D fragment: r=0..3 -> q2=r at ij2 = Mt*4 + half*2; r=4..7 -> ij2+1.
  // float4 store offset: ebase + ij2*512 + c*16 + p*4 (16B aligned).
  size_t st0 = ebase + (size_t)(half * 2) * 512 + c * 16 + p * 4;

  for (int Mt = 0; Mt < 9; ++Mt) {
    v2f afrag;
    afrag[0] = xin[baseA[0] + pix];
    afrag[1] = xin[baseA[1] + pix];

    v8f acc = {0.f, 0.f, 0.f, 0.f, 0.f, 0.f, 0.f, 0.f};
    acc = __builtin_amdgcn_wmma_f32_16x16x4_f32(
        false, afrag, false, bfrag, (short)0, acc, false, false);

    float4* e0 = reinterpret_cast<float4*>(E + st0);
    float4* e1 = reinterpret_cast<float4*>(E + st0 + 512);
    *e0 = make_float4(acc[0], acc[1], acc[2], acc[3]);
    *e1 = make_float4(acc[4], acc[5], acc[6], acc[7]);

    // advance row block: ij += 4  (j += 4 with carry into i)
    const int j4 = j + 4;
    if (j4 >= 6) { pix += 24; j = j4 - 6; }   // i+1 (+28), j-6 (-12), +8
    else         { pix += 8;  j = j4; }
    st0 += 4 * 512;                            // ij2 advances by 4 per Mt
  }
}

// ---------------------------------------------------------------------------
// Kernel 2: rh = max(R*aw, 0.01) precompute. One thread per (b,k,t).
// Decoded once per thread; output is L2-resident for the streaming pass.
// ---------------------------------------------------------------------------
__global__ __launch_bounds__(256) void rh_kernel(
    const float* __restrict__ xin, const float* __restrict__ R,
    float* __restrict__ RH, int useInit)
{
  const int idx = blockIdx.x * 256 + threadIdx.x;   // 2654208 = 8*288*1152
  const int b = idx / (NK * NT);
  const int rem = idx - b * (NK * NT);
  const int k = rem / NT;
  const int t = rem - k * NT;
  const int u = k / 9, r9 = k % 9;
  const int xk = r9 / 3, yk = r9 % 3;
  const int ij = t >> 5;
  const int i = ij / 6, j = ij % 6;
  const float aw =
      xin[((b * CH_X + 512 + u) * HH + 2 * i + xk) * HH + 2 * j + yk];
  const float Rv = useInit ? (1.0f / 1152.0f) : R[idx];
  RH[idx] = fmaxf(Rv * aw, 0.01f);
}

// ---------------------------------------------------------------------------
// Kernel 3: fused mu/sigma/sum_r. One thread per (b, f), reduction over k.
// Inner loop: two coalesced loads + 3 FMAs per k. sigma = E[V^2]-mu^2.
// ---------------------------------------------------------------------------
__global__ __launch_bounds__(256) void routing_stats(
    const float* __restrict__ E, const float* __restrict__ RH,
    float* __restrict__ MU, float* __restrict__ SG, float* __restrict__ SR)
{
  const int idx = blockIdx.x * 256 + threadIdx.x;   // 147456 = 8*18432
  const int b = idx / NF;
  const int f = idx - b * NF;
  const int t = f % NT;

  const size_t eb = (size_t)b * NK * NF + f;
  const int rb = (b * NK) * NT + t;

  float s0 = 0.f, s1 = 0.f, s2 = 0.f;
  for (int k = 0; k < NK; ++k) {
    const float rh = RH[rb + k * NT];
    const float e  = E[eb + (size_t)k * NF];
    if (k + 1 < NK) __builtin_prefetch(&E[eb + (size_t)(k + 1) * NF], 0, 1);
    s0 += rh;
    s1 += rh * e;
    s2 += rh * e * e;
  }
  const float mu = s1 / s0;
  const float sg = fmaxf(s2 / s0 - mu * mu, 0.01f);
  MU[idx] = mu;
  SG[idx] = sg;
  if (f < NT) SR[b * NT + f] = s0;   // one unique writer per (b,t)
}

// ---------------------------------------------------------------------------
// Kernel 4: activation + Gaussian normalizer product. One thread per (b,t).
// ---------------------------------------------------------------------------
__global__ __launch_bounds__(256) void act_kernel(
    const float* __restrict__ SG, const float* __restrict__ SR,
    const float* __restrict__ beta_v, const float* __restrict__ beta_a,
    const float* __restrict__ lam, float* __restrict__ ACT,
    float* __restrict__ PT)
{
  const int idx = blockIdx.x * 256 + threadIdx.x;   // 9216 = 8*1152
  const int b = idx / NT;
  const int t = idx - b * NT;
  const float sumr = SR[idx];
  const float bv = beta_v[0];
  const float L  = lam[0];
  float cost = 0.f, prod = 1.f;
#pragma unroll
  for (int s = 0; s < 16; ++s) {
    const float sg = SG[b * NF + s * NT + t];
    cost += bv + logf(sg);
    prod *= rsqrtf(6.28318530717958647692f * sg);
  }
  cost *= sumr;
  const float a = L * (beta_a[t & 31] - cost);
  ACT[idx] = 1.0f / (1.0f + expf(-a));
  PT[idx] = prod;
}

// ---------------------------------------------------------------------------
// Kernel 5: p_hat. One thread per (b,k,t). Reads 16 contiguous vote floats
// at the (perm[x],perm[y])-permuted k, compares to scrambled means MU.
// ---------------------------------------------------------------------------
__global__ __launch_bounds__(256) void phat_kernel(
    const float* __restrict__ E, const float* __restrict__ MU,
    const float* __restrict__ ACT, const float* __restrict__ PT,
    float* __restrict__ PH)
{
  const int idx = blockIdx.x * 256 + threadIdx.x;   // 2654208 = 8*288*1152
  const int b = idx / (NK * NT);
  const int rem = idx - b * (NK * NT);
  const int k = rem / NT;
  const int t = rem - k * NT;
  const int u = k / 9, r9 = k % 9;
  const int xk = r9 / 3, yk = r9 % 3;
  const int px = (xk == 0) ? 0 : ((xk == 1) ? 2 : 1);   // perm = [0,2,1]
  const int py = (yk == 0) ? 0 : ((yk == 1) ? 2 : 1);
  const int kp = u * 9 + px * 3 + py;

  const size_t ebase = (size_t)(b * NK + kp) * NF + (size_t)t * 16;
  const int mb = b * NF + t;
  float sum = 0.f;
#pragma unroll
  for (int pq = 0; pq < 16; ++pq) {
    const float d = E[ebase + pq] - MU[mb + pq * NT];
    sum += d * d;
  }
  PH[idx] = expf(-sum) * PT[b * NT + t] * ACT[b * NT + t];
}

// ---------------------------------------------------------------------------
// Kernel 6: deterministic scatter-sum S via gather (no float atomics).
// One thread per (b,u,h1,h2); <=2x2 covering windows, 32 contiguous c-reads.
// ---------------------------------------------------------------------------
__global__ __launch_bounds__(256) void sden_kernel(
    const float* __restrict__ PH, float* __restrict__ S)
{
  const int idx = blockIdx.x * 256 + threadIdx.x;   // 50176 = 8*32*196
  const int b = idx / (32 * HH * HH);
  const int rem = idx - b * (32 * HH * HH);
  const int u = rem / (HH * HH);
  const int hh = rem - u * (HH * HH);
  const int h1 = hh / HH, h2 = hh % HH;

  float s = 0.f;
  for (int xk = 0; xk < 3; ++xk) {
    const int ii = h1 - xk;
    if (ii < 0 || (ii & 1) || (ii >> 1) >= 6) continue;
    const int i = ii >> 1;
    for (int yk = 0; yk < 3; ++yk) {
      const int jj = h2 - yk;
      if (jj < 0 || (jj & 1) || (jj >> 1) >= 6) continue;
      const int j = jj >> 1;
      const int k = u * 9 + xk * 3 + yk;
      const int base = (b * NK + k) * NT + (i * 6 + j) * 32;
#pragma unroll
      for (int c = 0; c < 32; ++c) s += PH[base + c];
    }
  }
  S[idx] = s;
}

// ---------------------------------------------------------------------------
// Kernel 7: R update. One thread per (b,k,t).
// ---------------------------------------------------------------------------
__global__ __launch_bounds__(256) void rupdate_kernel(
    const float* __restrict__ PH, const float* __restrict__ S,
    float* __restrict__ R)
{
  const int idx = blockIdx.x * 256 + threadIdx.x;   // 2654208
  const int b = idx / (NK * NT);
  const int rem = idx - b * (NK * NT);
  const int k = rem / NT;
  const int t = rem - k * NT;
  const int u = k / 9, r9 = k % 9;
  const int xk = r9 / 3, yk = r9 % 3;
  const int ij = t >> 5;
  const int i = ij / 6, j = ij % 6;
  R[idx] = PH[idx] /
           S[((b * 32 + u) * HH + 2 * i + xk) * HH + 2 * j + yk];
}

// ---------------------------------------------------------------------------
// Kernel 8: final output assembly.
// out[b, c*16+s, i, j] = MU[b, s*1152 + (i*6+j)*32 + c]
// out[b, 512+c, i, j]  = ACT[b, (i*6+j)*32 + c]
// ---------------------------------------------------------------------------
__global__ __launch_bounds__(256) void out_kernel(
    const float* __restrict__ MU, const float* __restrict__ ACT,
    float* __restrict__ out)
{
  const int idx = blockIdx.x * 256 + threadIdx.x;   // 156672 = 8*544*36
  const int b = idx / (CH_X * 36);
  const int rem = idx - b * (CH_X * 36);
  const int ch = rem / 36;
  const int sp = rem - ch * 36;                     // i*6 + j
  if (ch < 512) {
    const int c = ch >> 4;
    const int s = ch & 15;
    out[idx] = MU[b * NF + s * NT + sp * 32 + c];
  } else {
    out[idx] = ACT[b * NT + sp * 32 + (ch - 512)];
  }
}

// ---------------------------------------------------------------------------
extern "C" void kernel_launch(void* const* d_in, const int* in_sizes, int n_in,
                              void* d_out, int out_size, void* d_ws,
                              size_t ws_size, hipStream_t stream)
{
  (void)in_sizes; (void)n_in; (void)out_size; (void)ws_size;

  const float* xin = (const float*)d_in[0];   // (8, 544, 14, 14)
  const float* lam = (const float*)d_in[1];   // scalar
  const float* Wt  = (const float*)d_in[2];   // (3,3,32,32,4,4)
  const float* bv  = (const float*)d_in[3];   // (1,)
  const float* ba  = (const float*)d_in[4];   // (32,)
  float* out = (float*)d_out;

  float* ws  = (float*)d_ws;
  float* E   = ws;                    // 42,467,328 floats
  float* R   = E   + (size_t)NB * NK * NF;
  float* RHP = R   + (size_t)NB * NK * NT;   // shared: RH then PH per iter
  float* MU  = RHP + (size_t)NB * NK * NT;
  float* SG  = MU  + (size_t)NB * NF;
  float* SR  = SG  + (size_t)NB * NF;
  float* ACT = SR  + (size_t)NB * NT;
  float* PT  = ACT + (size_t)NB * NT;
  float* S   = PT  + (size_t)NB * NT;

  capsvote_wmma<<<NB * NK, 256, 0, stream>>>(xin, Wt, E);

  for (int it = 0; it < 3; ++it) {
    rh_kernel<<<(NB * NK * NT) / 256, 256, 0, stream>>>(xin, R, RHP,
                                                        it == 0 ? 1 : 0);
    routing_stats<<<(NB * NF) / 256, 256, 0, stream>>>(E, RHP, MU, SG, SR);
    act_kernel<<<(NB * NT) / 256, 256, 0, stream>>>(SG, SR, bv, ba, lam,
                                                    ACT, PT);
    if (it < 2) {
      // RH is dead now; reuse the buffer for PH.
      phat_kernel<<<(NB * NK * NT) / 256, 256, 0, stream>>>(E, MU, ACT, PT,
                                                            RHP);
      sden_kernel<<<(NB * 32 * HH * HH) / 256, 256, 0, stream>>>(RHP, S);
      rupdate_kernel<<<(NB * NK * NT) / 256, 256, 0, stream>>>(RHP, S, R);
    }
  }

  out_kernel<<<(NB * CH_X * 36) / 256, 256, 0, stream>>>(MU, ACT, out);
}